// ConvPixelInceptionDecoder_2Layer_27023934226689
// MI455X (gfx1250) — compile-verified
//
#include <hip/hip_runtime.h>
#include <math.h>

// Problem constants (B=1)
#define HEADS 8
#define CDIM  128
#define HD    16          // head dim
#define HH    16
#define WW    32
#define DD    32
#define NTOT  (HH*WW*DD)  // 16384 spatial positions
#define ATTN_SCALE 0.25f  // (CDIM/HEADS)^-0.5 = 16^-0.5

typedef float v2f __attribute__((ext_vector_type(2)));
typedef float v8f __attribute__((ext_vector_type(8)));

// D(16x16,f32) = A(16x4,f32) x B(4x16,f32) + C  -- V_WMMA_F32_16X16X4_F32
static __device__ __forceinline__ v8f wmma4(v2f a, v2f b, v8f c) {
  return __builtin_amdgcn_wmma_f32_16x16x4_f32(false, a, false, b, (short)0, c,
                                               false, false);
}

// ---------------------------------------------------------------------------
// Async global->LDS copies (GLOBAL_LOAD_ASYNC_TO_LDS_B128, ASYNCcnt) when the
// toolchain exposes them; falls back to vectorized load+ds_store otherwise.
// Builtin signature (from diagnostics): (v4i addrspace(1)*, v4i addrspace(3)*,
// imm offset, imm cpol) where v4i = 16-byte int vector (b128 payload).
// ---------------------------------------------------------------------------
#if __has_builtin(__builtin_amdgcn_global_load_async_to_lds_b128) && \
    __has_builtin(__builtin_amdgcn_s_wait_asynccnt)
#define USE_ASYNC_LDS 1
typedef int v4i_ __attribute__((vector_size(4 * sizeof(int))));
typedef __attribute__((address_space(1))) v4i_ as1_v4i;
typedef __attribute__((address_space(3))) v4i_ as3_v4i;
#endif

static __device__ __forceinline__ void copy16_g2l(const float* g, float* l) {
#ifdef USE_ASYNC_LDS
  __builtin_amdgcn_global_load_async_to_lds_b128(
      (as1_v4i*)(uintptr_t)g,
      (as3_v4i*)(uint32_t)(uintptr_t)l,   // low 32 bits == LDS byte offset
      0, 0);
#else
  *(float4*)l = *(const float4*)g;
#endif
}

static __device__ __forceinline__ void async_join() {
#ifdef USE_ASYNC_LDS
  __builtin_amdgcn_s_wait_asynccnt(0);
#endif
}

// ---------------------------------------------------------------------------
// K1: q/k/v = xt @ {qw,kw,vw}.  xt[n,c] = x[c*N+n].  One wave per 16x16 tile,
// A-loads shared across the three accumulators.  Output row-major (N,C).
// ---------------------------------------------------------------------------
__global__ __launch_bounds__(32)
void qkv_gemm_kernel(const float* __restrict__ x,
                     const float* __restrict__ qw, const float* __restrict__ kw,
                     const float* __restrict__ vw,
                     float* __restrict__ qraw, float* __restrict__ kraw,
                     float* __restrict__ vraw) {
  const int n0 = blockIdx.x * 16;
  const int c0 = blockIdx.y * 16;
  const int lane = threadIdx.x & 31;
  const int lm = lane & 15;        // M / N column within tile
  const int lh = lane >> 4;        // lane-half selects K pair
  v8f aq = {0,0,0,0,0,0,0,0};
  v8f ak = {0,0,0,0,0,0,0,0};
  v8f av = {0,0,0,0,0,0,0,0};
  for (int k0 = 0; k0 < CDIM; k0 += 4) {
    const int ka = k0 + 2*lh;
    v2f a, b;
    // A[M=lm, K=ka..ka+1] = xt[n0+lm, ka] = x[ka*N + n0+lm]
    a.x = x[(size_t)(ka    )*NTOT + n0 + lm];
    a.y = x[(size_t)(ka + 1)*NTOT + n0 + lm];
    if (k0 + 4 < CDIM) __builtin_prefetch(&x[(size_t)(ka + 4)*NTOT + n0 + lm], 0, 1);
    b.x = qw[(ka    )*CDIM + c0 + lm];
    b.y = qw[(ka + 1)*CDIM + c0 + lm];
    aq = wmma4(a, b, aq);
    b.x = kw[(ka    )*CDIM + c0 + lm];
    b.y = kw[(ka + 1)*CDIM + c0 + lm];
    ak = wmma4(a, b, ak);
    b.x = vw[(ka    )*CDIM + c0 + lm];
    b.y = vw[(ka + 1)*CDIM + c0 + lm];
    av = wmma4(a, b, av);
  }
#pragma unroll
  for (int r = 0; r < 8; ++r) {
    const int n = n0 + r + 8*lh;   // D layout: VGPR r, lane-half -> M
    const int c = c0 + lm;
    qraw[(size_t)n*CDIM + c] = aq[r];
    kraw[(size_t)n*CDIM + c] = ak[r];
    vraw[(size_t)n*CDIM + c] = av[r];
  }
}

// ---------------------------------------------------------------------------
// K2: softmax of q over head-dim (16 contiguous cols) + transpose to [c][n]
// ---------------------------------------------------------------------------
__global__ void softmax_q_kernel(const float* __restrict__ qraw,
                                 float* __restrict__ q3) {
  const int tid = blockIdx.x * blockDim.x + threadIdx.x;
  const int head = tid >> 14;          // / NTOT
  const int n    = tid & (NTOT - 1);
  float v[16];
  float m = -3.0e38f;
#pragma unroll
  for (int j = 0; j < 16; ++j) {
    v[j] = qraw[(size_t)n*CDIM + head*16 + j];
    m = fmaxf(m, v[j]);
  }
  float s = 0.f;
#pragma unroll
  for (int j = 0; j < 16; ++j) { v[j] = __expf(v[j] - m); s += v[j]; }
  const float inv = 1.f / s;
#pragma unroll
  for (int j = 0; j < 16; ++j)
    q3[(size_t)(head*16 + j)*NTOT + n] = v[j] * inv;
}

// ---------------------------------------------------------------------------
// K3: softmax of k over the HEADS axis (cols j, j+16, ..., j+112) + transpose
// ---------------------------------------------------------------------------
__global__ void softmax_k_kernel(const float* __restrict__ kraw,
                                 float* __restrict__ k3) {
  const int tid = blockIdx.x * blockDim.x + threadIdx.x;
  const int j = tid >> 14;             // 0..15
  const int n = tid & (NTOT - 1);
  float v[8];
  float m = -3.0e38f;
#pragma unroll
  for (int h = 0; h < 8; ++h) {
    v[h] = kraw[(size_t)n*CDIM + h*16 + j];
    m = fmaxf(m, v[h]);
  }
  float s = 0.f;
#pragma unroll
  for (int h = 0; h < 8; ++h) { v[h] = __expf(v[h] - m); s += v[h]; }
  const float inv = 1.f / s;
#pragma unroll
  for (int h = 0; h < 8; ++h)
    k3[(size_t)(h*16 + j)*NTOT + n] = v[h] * inv;
}

// ---------------------------------------------------------------------------
// K4: transpose v (N,C) -> (C,N)
// ---------------------------------------------------------------------------
__global__ void transpose_v_kernel(const float* __restrict__ vraw,
                                   float* __restrict__ v3) {
  const int tid = blockIdx.x * blockDim.x + threadIdx.x;   // = c*NTOT + n
  const int c = tid >> 14;
  const int n = tid & (NTOT - 1);
  v3[tid] = vraw[(size_t)n*CDIM + c];
}

// ---------------------------------------------------------------------------
// K5: inclusive 3-D summed-area table over contiguous (16,32,32) volumes.
// One block per volume; whole volume staged in 64 KB of LDS via async b128
// copies (this kernel streams the 128 MB ikv tensor => dominant data mover).
// ---------------------------------------------------------------------------
__global__ __launch_bounds__(256)
void sat3d_kernel(const float* __restrict__ src, float* __restrict__ dst) {
  __shared__ float vol[NTOT];
  const float* s = src + (size_t)blockIdx.x * NTOT;
  float* dptr = dst + (size_t)blockIdx.x * NTOT;
  for (int t4 = threadIdx.x * 4; t4 < NTOT; t4 += 256 * 4)
    copy16_g2l(s + t4, vol + t4);
  async_join();
  __syncthreads();
  // cumsum along d: 512 (h,w)-rows of 32
  for (int r = threadIdx.x; r < 512; r += 256) {
    const int base = r * 32;
    float run = 0.f;
    for (int d = 0; d < 32; ++d) { run += vol[base + d]; vol[base + d] = run; }
  }
  __syncthreads();
  // cumsum along w: 512 (h,d)-columns
  for (int r = threadIdx.x; r < 512; r += 256) {
    const int h = r >> 5, d = r & 31;
    float run = 0.f;
    for (int w = 0; w < 32; ++w) {
      const int idx = h*1024 + w*32 + d;
      run += vol[idx]; vol[idx] = run;
    }
  }
  __syncthreads();
  // cumsum along h: 1024 (w,d)-columns
  for (int r = threadIdx.x; r < 1024; r += 256) {
    float run = 0.f;
    for (int h = 0; h < 16; ++h) {
      const int idx = h*1024 + r;
      run += vol[idx]; vol[idx] = run;
    }
  }
  __syncthreads();
  for (int t4 = threadIdx.x * 4; t4 < NTOT; t4 += 256 * 4)
    *(float4*)(dptr + t4) = *(const float4*)(vol + t4);
}

// ---------------------------------------------------------------------------
// K6: kv[bh,i,j,h,:,:] = k3slice_i(32x32) @ v3slice_j(32x32) for all 16x16
// (i,j).  One block per (h,bh).  Both operand families staged in LDS (async
// b128 fills); rows padded to stride 36 floats: 16B-aligned and 36*lane mod 64
// hits 16 distinct banks => conflict-free A reads.  Each wave: 32 pairs x
// 4 tiles x 8 K-steps of V_WMMA_F32_16X16X4_F32.
// ---------------------------------------------------------------------------
#define LSTR  36           // padded LDS row stride (floats)
#define SLICE (32*LSTR)    // 1152 floats per 32x32 slice

__global__ __launch_bounds__(256)
void kv_gemm_kernel(const float* __restrict__ k3, const float* __restrict__ v3,
                    float* __restrict__ kv) {
  __shared__ float Alds[16 * SLICE];   // 72 KB: A_i[w][m]
  __shared__ float Blds[16 * SLICE];   // 72 KB: B_j[m][d]
  const int h  = blockIdx.x;            // 0..15
  const int bh = blockIdx.y;            // 0..7
  // 16 slices * 32 rows * 8 four-float chunks = 4096 chunks per family
  for (int t = threadIdx.x; t < 4096; t += 256) {
    const int i   = t >> 8;            // slice
    const int row = (t >> 3) & 31;     // row within slice
    const int ch  = (t & 7) << 2;      // float offset of 16-byte chunk
    const size_t gofs = (size_t)(bh*16 + i)*NTOT + h*1024 + row*32 + ch;
    const int    lofs = i*SLICE + row*LSTR + ch;
    copy16_g2l(k3 + gofs, Alds + lofs);
    copy16_g2l(v3 + gofs, Blds + lofs);
  }
  async_join();
  __syncthreads();
  const int wave = threadIdx.x >> 5;
  const int lane = threadIdx.x & 31;
  const int lm = lane & 15, lh = lane >> 4;
  for (int p = wave; p < 256; p += 8) {
    const int i = p >> 4, j = p & 15;
    const float* A = Alds + i*SLICE;                              // [w][m]
    const float* B = Blds + j*SLICE;                              // [m][d]
    float* out = kv + ((size_t)((bh*16 + i)*16 + j))*NTOT + h*1024;
#pragma unroll
    for (int mt = 0; mt < 32; mt += 16) {
#pragma unroll
      for (int nt = 0; nt < 32; nt += 16) {
        v8f acc = {0,0,0,0,0,0,0,0};
#pragma unroll
        for (int k0 = 0; k0 < 32; k0 += 4) {
          const int kk = k0 + 2*lh;
          v2f a, b;
          a.x = A[(mt + lm)*LSTR + kk];
          a.y = A[(mt + lm)*LSTR + kk + 1];
          b.x = B[(kk    )*LSTR + nt + lm];
          b.y = B[(kk + 1)*LSTR + nt + lm];
          acc = wmma4(a, b, acc);
        }
#pragma unroll
        for (int r = 0; r < 8; ++r) {
          const int w = mt + r + 8*lh, d = nt + lm;
          out[w*32 + d] = acc[r];
        }
      }
    }
  }
}

// ---------------------------------------------------------------------------
// K7: windowed box sums via 8-corner SAT differences (edge-clamped center
// padding), contracted with q; accumulates 0.5 * sum over both windows of
// SCALE*num/Z into acc[c][n].
// ---------------------------------------------------------------------------
__global__ __launch_bounds__(256)
void wkv_num_kernel(const float* __restrict__ q3, const float* __restrict__ ik,
                    const float* __restrict__ ikv, float* __restrict__ accout) {
  const int bh = blockIdx.y;
  const int voxel = blockIdx.x * 256 + threadIdx.x;
  const int h = voxel >> 10, w = (voxel >> 5) & 31, d = voxel & 31;
  float qv[16];
#pragma unroll
  for (int i = 0; i < 16; ++i)
    qv[i] = q3[(size_t)(bh*16 + i)*NTOT + voxel];
  float res[16];
#pragma unroll
  for (int j = 0; j < 16; ++j) res[j] = 0.f;

#pragma unroll 1
  for (int win = 0; win < 2; ++win) {
    const int a = win ? 8 : 4;
    const int da = a - 1;
    const int oh = HH - a + 1, ow = WW - a + 1, od = DD - a + 1;
    const int ph = (HH - oh) >> 1, pw = (WW - ow) >> 1, pd = (DD - od) >> 1;
    int th = h - ph; th = th < 0 ? 0 : (th > oh - 1 ? oh - 1 : th);
    int tw = w - pw; tw = tw < 0 ? 0 : (tw > ow - 1 ? ow - 1 : tw);
    int td = d - pd; td = td < 0 ? 0 : (td > od - 1 ? od - 1 : td);
    const int h1 = th + da, w1 = tw + da, d1 = td + da;
    int off[8]; float sgn[8];
    off[0] = h1*1024 + w1*32 + d1; sgn[0] =  1.f;
    off[1] = h1*1024 + w1*32 + td; sgn[1] = -1.f;
    off[2] = h1*1024 + tw*32 + d1; sgn[2] = -1.f;
    off[3] = h1*1024 + tw*32 + td; sgn[3] =  1.f;
    off[4] = th*1024 + w1*32 + d1; sgn[4] = -1.f;
    off[5] = th*1024 + w1*32 + td; sgn[5] =  1.f;
    off[6] = th*1024 + tw*32 + d1; sgn[6] =  1.f;
    off[7] = th*1024 + tw*32 + td; sgn[7] = -1.f;

    float Z = 1e-6f;
#pragma unroll 1
    for (int i = 0; i < 16; ++i) {
      const float* vv = ik + (size_t)(bh*16 + i)*NTOT;
      float wk = 0.f;
#pragma unroll
      for (int c = 0; c < 8; ++c) wk += sgn[c] * vv[off[c]];
      Z += qv[i] * wk;
    }
    float num[16];
#pragma unroll
    for (int j = 0; j < 16; ++j) num[j] = 0.f;
#pragma unroll 1
    for (int i = 0; i < 16; ++i) {
      const float qi = qv[i];
#pragma unroll 1
      for (int j = 0; j < 16; ++j) {
        const float* vv = ikv + ((size_t)((bh*16 + i)*16 + j))*NTOT;
        float s8 = 0.f;
#pragma unroll
        for (int c = 0; c < 8; ++c) s8 += sgn[c] * vv[off[c]];
        num[j] += qi * s8;
      }
    }
    const float f = ATTN_SCALE / Z;
#pragma unroll
    for (int j = 0; j < 16; ++j) res[j] += num[j] * f;
  }
#pragma unroll
  for (int j = 0; j < 16; ++j)
    accout[(size_t)(bh*16 + j)*NTOT + voxel] = 0.5f * res[j];   // / len(WINDOWS)
}

// ---------------------------------------------------------------------------
// K8: final projection: d_out[c][n] = sum_c' acc[c'][n]*pw[c'][c] + pb[c]
// ---------------------------------------------------------------------------
__global__ __launch_bounds__(32)
void proj_gemm_kernel(const float* __restrict__ accb,
                      const float* __restrict__ pw,
                      const float* __restrict__ pb,
                      float* __restrict__ out) {
  const int n0 = blockIdx.x * 16;
  const int c0 = blockIdx.y * 16;
  const int lane = threadIdx.x & 31;
  const int lm = lane & 15, lh = lane >> 4;
  v8f acc = {0,0,0,0,0,0,0,0};
  for (int k0 = 0; k0 < CDIM; k0 += 4) {
    const int ka = k0 + 2*lh;
    v2f a, b;
    a.x = accb[(size_t)(ka    )*NTOT + n0 + lm];
    a.y = accb[(size_t)(ka + 1)*NTOT + n0 + lm];
    b.x = pw[(ka    )*CDIM + c0 + lm];
    b.y = pw[(ka + 1)*CDIM + c0 + lm];
    acc = wmma4(a, b, acc);
  }
  const float bias = pb[c0 + lm];
#pragma unroll
  for (int r = 0; r < 8; ++r) {
    const int n = n0 + r + 8*lh;
    const int c = c0 + lm;
    out[(size_t)c*NTOT + n] = acc[r] + bias;   // transpose back to (C,N)
  }
}

// ---------------------------------------------------------------------------
extern "C" void kernel_launch(void* const* d_in, const int* in_sizes, int n_in,
                              void* d_out, int out_size, void* d_ws, size_t ws_size,
                              hipStream_t stream) {
  const float* x  = (const float*)d_in[0];
  const float* qw = (const float*)d_in[1];
  const float* kw = (const float*)d_in[2];
  const float* vw = (const float*)d_in[3];
  const float* pw = (const float*)d_in[4];
  const float* pb = (const float*)d_in[5];
  float* out = (float*)d_out;

  const size_t U = (size_t)NTOT * CDIM;     // 2,097,152 floats
  float* ws   = (float*)d_ws;
  float* qraw = ws;                         // (N,C)
  float* kraw = ws + 1*U;
  float* vraw = ws + 2*U;
  float* q3   = ws + 3*U;                   // (C,N)
  float* k3   = ws + 4*U;
  float* v3   = ws + 5*U;
  float* ik   = ws + 6*U;                   // SAT of k3
  float* accb = ws + 7*U;                   // attention output (C,N)
  float* ikv  = ws + 8*U;                   // kv then SAT(kv): 16*U floats

  qkv_gemm_kernel<<<dim3(NTOT/16, CDIM/16), 32, 0, stream>>>(x, qw, kw, vw,
                                                             qraw, kraw, vraw);
  softmax_q_kernel<<<(NTOT*HEADS)/256, 256, 0, stream>>>(qraw, q3);
  softmax_k_kernel<<<(NTOT*HD)/256, 256, 0, stream>>>(kraw, k3);
  transpose_v_kernel<<<(NTOT*CDIM)/256, 256, 0, stream>>>(vraw, v3);
  sat3d_kernel<<<CDIM, 256, 0, stream>>>(k3, ik);
  kv_gemm_kernel<<<dim3(HH, HEADS), 256, 0, stream>>>(k3, v3, ikv);
  sat3d_kernel<<<CDIM*HD, 256, 0, stream>>>(ikv, ikv);
  wkv_num_kernel<<<dim3(NTOT/256, HEADS), 256, 0, stream>>>(q3, ik, ikv, accb);
  proj_gemm_kernel<<<dim3(NTOT/16, CDIM/16), 32, 0, stream>>>(accb, pw, pb, out);
}